// GNNModel_31456340476380
// MI455X (gfx1250) — compile-verified
//
#include <hip/hip_runtime.h>

// ---------------------------------------------------------------------------
// 2-layer GCN (PyG GCNConv semantics, add_self_loops=False) for MI455X gfx1250.
// Key algebraic rewrite:  out[d] = dinv[d] * sum_{e:(s->d)} dinv[s]*h[s]
// so the per-edge work is a pure gather + fp32 atomic scatter-add; dinv/bias/
// relu are applied in cheap elementwise epilogues. GEMMs (N x 64 @ 64 x 64)
// use fp32 WMMA V_WMMA_F32_16X16X4_F32 (exact fp32, K=4 per issue).
// Whole working set (~130 MB) fits in the 192 MB L2 -> scatter atomics are
// L2-local; this kernel is L2-bandwidth/atomic bound, not HBM bound.
// ---------------------------------------------------------------------------

typedef __attribute__((ext_vector_type(2))) float v2f;
typedef __attribute__((ext_vector_type(8))) float v8f;

#define HDIM      64
#define LDS_PITCH 66   // even pad: keeps ds_load_b64 8B-aligned, no bank conflicts

// ---------------- simple fills / degree / dinv ----------------

__global__ void k_zero_f32(float* __restrict__ p, long long n) {
  long long i = (long long)blockIdx.x * blockDim.x + threadIdx.x;
  if (i < n) p[i] = 0.0f;
}

__global__ void k_degree(const long long* __restrict__ dst, float* __restrict__ deg, int E) {
  int e = blockIdx.x * blockDim.x + threadIdx.x;
  if (e < E) atomicAdd(&deg[(int)dst[e]], 1.0f);
}

__global__ void k_dinv(float* __restrict__ degdinv, int N) {
  int i = blockIdx.x * blockDim.x + threadIdx.x;
  if (i < N) {
    float dg = degdinv[i];
    degdinv[i] = (dg > 0.0f) ? rsqrtf(fmaxf(dg, 1.0f)) : 0.0f;
  }
}

// ---------------- fp32 WMMA GEMM: Y[r,:] = dinv[r] * (X[rowp(r),:] @ W) -----
// One wave computes a 16x64 output stripe: 4 accumulators of 16x16, K=64 via
// 16 x V_WMMA_F32_16X16X4_F32 per accumulator (64 WMMA per wave).
// A frag (16x4 f32): lane l<16 -> row l, k={4t,4t+1}; lane l>=16 -> k={4t+2,4t+3}.
// B frag (4x16 f32): mirrored; read from LDS-transposed W so each frag is one
// aligned ds_load_b64.  C/D: VGPR i -> row i (lanes 0-15) / i+8 (lanes 16-31).

__global__ __launch_bounds__(256)
void k_gemm64_scale(const float* __restrict__ X,
                    const long long* __restrict__ idx,   // optional row gather (node_ids)
                    const float* __restrict__ W,         // [64][64] row-major (k, n)
                    const float* __restrict__ dinv,
                    float* __restrict__ Y, int N) {
  __shared__ float wl[HDIM * LDS_PITCH];

  // Stage W transposed: wl[n*66 + k] = W[k*64 + n]. Write banks = 2*lane -> clean.
  for (int i = threadIdx.x; i < HDIM * HDIM; i += 256) {
    int k = i >> 6, n = i & 63;
    wl[n * LDS_PITCH + k] = W[i];
  }
  __syncthreads();

  const int wave = threadIdx.x >> 5;
  const int lane = threadIdx.x & 31;
  const int r0   = (blockIdx.x * 8 + wave) * 16;
  if (r0 + 16 > N) return;                 // wave-uniform: EXEC all-ones for WMMA

  const int m    = lane & 15;              // row (A) / col (B) within tile
  const int koff = (lane >> 4) * 2;        // half-wave K offset

  const long long rowp = idx ? idx[r0 + m] : (long long)(r0 + m);
  const float* xrow = X + rowp * HDIM + koff;

  const float* w0 = &wl[( 0 + m) * LDS_PITCH + koff];
  const float* w1 = &wl[(16 + m) * LDS_PITCH + koff];
  const float* w2 = &wl[(32 + m) * LDS_PITCH + koff];
  const float* w3 = &wl[(48 + m) * LDS_PITCH + koff];

  v8f acc0 = {}, acc1 = {}, acc2 = {}, acc3 = {};

#pragma unroll
  for (int t = 0; t < 16; ++t) {
    const v2f a  = *(const v2f*)(xrow + 4 * t);
    v2f b;
    b = *(const v2f*)(w0 + 4 * t);
    acc0 = __builtin_amdgcn_wmma_f32_16x16x4_f32(false, a, false, b, (short)0, acc0, false, false);
    b = *(const v2f*)(w1 + 4 * t);
    acc1 = __builtin_amdgcn_wmma_f32_16x16x4_f32(false, a, false, b, (short)0, acc1, false, false);
    b = *(const v2f*)(w2 + 4 * t);
    acc2 = __builtin_amdgcn_wmma_f32_16x16x4_f32(false, a, false, b, (short)0, acc2, false, false);
    b = *(const v2f*)(w3 + 4 * t);
    acc3 = __builtin_amdgcn_wmma_f32_16x16x4_f32(false, a, false, b, (short)0, acc3, false, false);
  }

  // Epilogue: scale rows by dinv and store (pre-scaled messages for scatter).
  const int rbase = r0 + ((lane >> 4) << 3);
  float dv[8];
#pragma unroll
  for (int i = 0; i < 8; ++i) dv[i] = dinv[rbase + i];
#pragma unroll
  for (int i = 0; i < 8; ++i) {
    const long long ro = (long long)(rbase + i) * HDIM;
    Y[ro +  0 + m] = dv[i] * acc0[i];
    Y[ro + 16 + m] = dv[i] * acc1[i];
    Y[ro + 32 + m] = dv[i] * acc2[i];
    Y[ro + 48 + m] = dv[i] * acc3[i];
  }
}

// ---------------- edge scatter: acc[dst,:] += hs[src,:] --------------------
// One edge per half-wave: 16 lanes x float4 = coalesced 256B row read,
// 4 contiguous global_atomic_add_f32 per lane.

__global__ __launch_bounds__(256)
void k_scatter(const long long* __restrict__ src, const long long* __restrict__ dst,
               const float* __restrict__ hs, float* __restrict__ acc, int E) {
  const long long tid  = (long long)blockIdx.x * blockDim.x + threadIdx.x;
  const int lane = threadIdx.x & 31;
  const long long e = (tid >> 5) * 2 + (lane >> 4);
  if (e >= E) return;
  const long long s = src[e];
  const long long d = dst[e];
  const int c = (lane & 15) * 4;
  const float4 v = *(const float4*)&hs[s * HDIM + c];
  float* p = &acc[d * HDIM + c];
  atomicAdd(p + 0, v.x);
  atomicAdd(p + 1, v.y);
  atomicAdd(p + 2, v.z);
  atomicAdd(p + 3, v.w);
}

// ---------------- epilogue: out = (relu?)(dinv[row]*acc + bias[col]) -------

__global__ void k_bias_act(const float* __restrict__ acc, const float* __restrict__ dinv,
                           const float* __restrict__ bias, float* __restrict__ out,
                           long long total, int relu) {
  long long i = (long long)blockIdx.x * blockDim.x + threadIdx.x;
  if (i >= total) return;
  const int row = (int)(i >> 6);
  const int col = (int)(i & 63);
  float v = dinv[row] * acc[i] + bias[col];
  out[i] = relu ? fmaxf(v, 0.0f) : v;
}

// ---------------------------------------------------------------------------

static inline size_t align256(size_t x) { return (x + 255) & ~(size_t)255; }

extern "C" void kernel_launch(void* const* d_in, const int* in_sizes, int n_in,
                              void* d_out, int out_size, void* d_ws, size_t ws_size,
                              hipStream_t stream) {
  const long long* node_ids = (const long long*)d_in[0];
  const long long* ei       = (const long long*)d_in[1];   // [2][E] int64
  const float*     emb      = (const float*)d_in[2];       // [N][64]
  const float*     W1       = (const float*)d_in[3];       // [64][64]
  const float*     b1       = (const float*)d_in[4];       // [64]
  const float*     W2       = (const float*)d_in[5];       // [64][64]
  const float*     b2       = (const float*)d_in[6];       // [64]
  float*           out      = (float*)d_out;               // [N][64]

  const int N = in_sizes[0];
  const int E = in_sizes[1] / 2;
  const long long NH = (long long)N * HDIM;

  const long long* src = ei;
  const long long* dst = ei + E;

  // Workspace carve-up: dinv (N f32) + A (N*64) + B (N*64) ~= 52 MB.
  char*  ws   = (char*)d_ws;
  float* dinv = (float*)ws;
  float* bufA = (float*)(ws + align256((size_t)N * sizeof(float)));
  float* bufB = bufA + NH;
  (void)ws_size; (void)n_in; (void)out_size;

  const int T = 256;
  const int gN   = (N + T - 1) / T;
  const int gE   = (E + T - 1) / T;
  const int gNH  = (int)((NH + T - 1) / T);
  const int gGemm = ((N + 15) / 16 + 7) / 8;                 // 8 waves/block, 16 rows/wave
  const int gScat = (int)((((long long)(E + 1) / 2) * 32 + T - 1) / T);

  // degree -> dinv (shared by both convs: same edge_index)
  k_zero_f32<<<gN, T, 0, stream>>>(dinv, N);
  k_degree  <<<gE, T, 0, stream>>>(dst, dinv, E);
  k_dinv    <<<gN, T, 0, stream>>>(dinv, N);

  // ---- conv1 ----
  k_gemm64_scale<<<gGemm, T, 0, stream>>>(emb, node_ids, W1, dinv, bufA, N); // A = dinv*(emb@W1)
  k_zero_f32   <<<gNH, T, 0, stream>>>(bufB, NH);
  k_scatter    <<<gScat, T, 0, stream>>>(src, dst, bufA, bufB, E);           // B += A[src]
  k_bias_act   <<<gNH, T, 0, stream>>>(bufB, dinv, b1, bufA, NH, 1);         // A = relu(dinv*B+b1)

  // ---- conv2 ---- (d_out used as scratch for pre-scatter messages)
  k_gemm64_scale<<<gGemm, T, 0, stream>>>(bufA, nullptr, W2, dinv, out, N);  // out = dinv*(A@W2)
  k_zero_f32   <<<gNH, T, 0, stream>>>(bufB, NH);
  k_scatter    <<<gScat, T, 0, stream>>>(src, dst, out, bufB, E);            // B += out[src]
  k_bias_act   <<<gNH, T, 0, stream>>>(bufB, dinv, b2, out, NH, 0);          // out = dinv*B+b2
}